// RWKV_RNN_9268539425093
// MI455X (gfx1250) — compile-verified
//
#include <hip/hip_runtime.h>
#include <math.h>

// RWKV single-token forward for gfx1250 (MI455X).
// Bandwidth-bound (~860MB fp32 weights / token); matvecs run through
// V_WMMA_F32_16X16X4_F32 with a broadcast-B trick (16 rows per wave).

#define L_  12
#define E_  1024
#define H_  16
#define S_  64
#define FF_ 3584
#define V_  50304
#define SCALE_  (1.0f / 128.0f)
#define GN_EPS_ 0.00064f

typedef __attribute__((ext_vector_type(2))) float v2f;
typedef __attribute__((ext_vector_type(8))) float v8f;

// ---------------- reductions (wave32) ----------------
__device__ __forceinline__ float warp_sum(float v) {
#pragma unroll
  for (int o = 16; o > 0; o >>= 1) v += __shfl_xor(v, o, 32);
  return v;
}

// blockDim must be 256 (8 waves)
__device__ __forceinline__ float block_sum256(float v, float* s8) {
  v = warp_sum(v);
  const int w = threadIdx.x >> 5;
  if ((threadIdx.x & 31) == 0) s8[w] = v;
  __syncthreads();
  float t = (threadIdx.x < 8) ? s8[threadIdx.x] : 0.0f;
  t = warp_sum(t);
  if (threadIdx.x == 0) s8[0] = t;
  __syncthreads();
  float r = s8[0];
  __syncthreads();
  return r;
}

// ---------------- WMMA matvec: y[m] = epilogue(sum_k W[m,k] * x[k]) --------
// One wave owns a 16-row tile. A-layout (ISA 16x4 fp32): lane<16 -> K={k0,k0+1},
// lane>=16 -> K={k0+2,k0+3}, row = lane&15. B replicates x in all 16 columns.
// modes: 0 plain (+addv if set), 1 silu, 2 sigmoid, 3 relu^2, 4 addv + mulv*dot
__global__ __launch_bounds__(256) void matvec_wmma(
    const float* __restrict__ W, const float* __restrict__ x,
    float* __restrict__ y, const float* __restrict__ addv,
    const float* __restrict__ mulv, int M, int K, int mode) {
  const int lane = threadIdx.x & 31;
  const int wave = threadIdx.x >> 5;
  const int row0 = (blockIdx.x * 8 + wave) * 16;
  if (row0 >= M) return;

  const int r    = lane & 15;
  const int koff = (lane >> 4) << 1;  // 0 or 2
  const float* wp = W + (size_t)(row0 + r) * (size_t)K + koff;
  const float* xp = x + koff;

  v8f acc = {};
#pragma unroll 4
  for (int k0 = 0; k0 < K; k0 += 4) {
    v2f a = *(const v2f*)(wp + k0);   // 256B of W per wave per WMMA
    v2f b = *(const v2f*)(xp + k0);   // L0-resident activation
    acc = __builtin_amdgcn_wmma_f32_16x16x4_f32(
        false, a, false, b, (short)0, acc, false, false);
  }

  // D layout: acc[j] holds row j (lanes 0-15) / row j+8 (lanes 16-31); all
  // 16 columns identical, so lane 0 and lane 16 each emit 8 rows.
  if (lane == 0 || lane == 16) {
    const int rbase = row0 + ((lane == 0) ? 0 : 8);
#pragma unroll
    for (int j = 0; j < 8; ++j) {
      float d = acc[j];
      const int m = rbase + j;
      float o;
      if (mode == 1) {                      // silu
        o = d / (1.0f + expf(-d)) ;
      } else if (mode == 2) {               // sigmoid
        o = 1.0f / (1.0f + expf(-d));
      } else if (mode == 3) {               // relu^2
        float t = fmaxf(d, 0.0f); o = t * t;
      } else if (mode == 4) {               // residual + gate
        o = addv[m] + mulv[m] * d;
      } else {
        o = addv ? (addv[m] + d) : d;
      }
      y[m] = o;
    }
  }
}

// ---------------- embedding + ln0 (1 block) ----------------
__global__ __launch_bounds__(256) void embed_ln(
    const int* __restrict__ token, const float* __restrict__ emb_w,
    const float* __restrict__ lw, const float* __restrict__ lb,
    float* __restrict__ x) {
  __shared__ float s8[8];
  const float* row = emb_w + (size_t)token[0] * E_;
  float lv[4]; float sum = 0.0f;
#pragma unroll
  for (int j = 0; j < 4; ++j) { lv[j] = row[threadIdx.x + j * 256]; sum += lv[j]; }
  const float mean = block_sum256(sum, s8) * (1.0f / E_);
  float vs = 0.0f;
#pragma unroll
  for (int j = 0; j < 4; ++j) { float t = lv[j] - mean; vs += t * t; }
  const float rstd = rsqrtf(block_sum256(vs, s8) * (1.0f / E_) + 1e-5f);
#pragma unroll
  for (int j = 0; j < 4; ++j) {
    const int i = threadIdx.x + j * 256;
    x[i] = (lv[j] - mean) * rstd * lw[i] + lb[i];
  }
}

// ---------------- att pre: ln1 + token-shift mixes (1 block) ----------------
__global__ __launch_bounds__(256) void att_pre(
    const float* __restrict__ x, const float* __restrict__ lw,
    const float* __restrict__ lb, const float* __restrict__ s1,
    const float* __restrict__ tmk, const float* __restrict__ tmv,
    const float* __restrict__ tmr, const float* __restrict__ tmg,
    float* __restrict__ xk, float* __restrict__ xv,
    float* __restrict__ xr, float* __restrict__ xg,
    float* __restrict__ new_as) {
  __shared__ float s8[8];
  float lv[4]; float sum = 0.0f;
#pragma unroll
  for (int j = 0; j < 4; ++j) { lv[j] = x[threadIdx.x + j * 256]; sum += lv[j]; }
  const float mean = block_sum256(sum, s8) * (1.0f / E_);
  float vs = 0.0f;
#pragma unroll
  for (int j = 0; j < 4; ++j) { float t = lv[j] - mean; vs += t * t; }
  const float rstd = rsqrtf(block_sum256(vs, s8) * (1.0f / E_) + 1e-5f);
#pragma unroll
  for (int j = 0; j < 4; ++j) {
    const int i = threadIdx.x + j * 256;
    const float xln = (lv[j] - mean) * rstd * lw[i] + lb[i];
    const float s = s1[i];
    xk[i] = xln * tmk[i] + s * (1.0f - tmk[i]);
    xv[i] = xln * tmv[i] + s * (1.0f - tmv[i]);
    xr[i] = xln * tmr[i] + s * (1.0f - tmr[i]);
    xg[i] = xln * tmg[i] + s * (1.0f - tmg[i]);
    new_as[i] = xln;
  }
}

// ---------------- ffn pre: ln2 + token-shift mixes (1 block) ----------------
__global__ __launch_bounds__(256) void ffn_pre(
    const float* __restrict__ x, const float* __restrict__ lw,
    const float* __restrict__ lb, const float* __restrict__ s3,
    const float* __restrict__ tmk, const float* __restrict__ tmr,
    float* __restrict__ xk2, float* __restrict__ xr2,
    float* __restrict__ new_fs) {
  __shared__ float s8[8];
  float lv[4]; float sum = 0.0f;
#pragma unroll
  for (int j = 0; j < 4; ++j) { lv[j] = x[threadIdx.x + j * 256]; sum += lv[j]; }
  const float mean = block_sum256(sum, s8) * (1.0f / E_);
  float vs = 0.0f;
#pragma unroll
  for (int j = 0; j < 4; ++j) { float t = lv[j] - mean; vs += t * t; }
  const float rstd = rsqrtf(block_sum256(vs, s8) * (1.0f / E_) + 1e-5f);
#pragma unroll
  for (int j = 0; j < 4; ++j) {
    const int i = threadIdx.x + j * 256;
    const float xln = (lv[j] - mean) * rstd * lw[i] + lb[i];
    const float s = s3[i];
    xk2[i] = xln * tmk[i] + s * (1.0f - tmk[i]);
    xr2[i] = xln * tmr[i] + s * (1.0f - tmr[i]);
    new_fs[i] = xln;
  }
}

// ---------------- per-head wkv + state update + groupnorm + gate ------------
// one block per head; 256 threads: t = tid&63 (column), sg = tid>>6 (s chunk)
__global__ __launch_bounds__(256) void wkv_gn(
    const float* __restrict__ rv, const float* __restrict__ kv,
    const float* __restrict__ vv, const float* __restrict__ gv,
    const float* __restrict__ tf, const float* __restrict__ td,
    const float* __restrict__ st_old, const float* __restrict__ lnxw,
    const float* __restrict__ lnxb, float* __restrict__ st_new,
    float* __restrict__ yg) {
  const int h = blockIdx.x;
  const int t = threadIdx.x & 63;
  const int sg = threadIdx.x >> 6;  // 0..3
  __shared__ float sh_r[64], sh_k[64], sh_v[64], sh_tf[64], sh_td[64];
  __shared__ float part[4][64];
  __shared__ float wkv_s[64];
  __shared__ float red[8];

  const float* ob = st_old + (size_t)h * S_ * S_;
  float* nb = st_new + (size_t)h * S_ * S_;
  if (threadIdx.x < 64) {
    sh_r[t]  = rv[h * S_ + t];
    sh_k[t]  = kv[h * S_ + t];
    sh_v[t]  = vv[h * S_ + t];
    sh_tf[t] = tf[h * S_ + t];
    sh_td[t] = td[h * S_ + t];
  }
  __syncthreads();

  // dotA = sum_s r[s]*tf[s]*k[s]
  float da = (threadIdx.x < 64) ? (sh_r[t] * sh_tf[t] * sh_k[t]) : 0.0f;
  da = block_sum256(da, red);

  // rs[t] = sum_s r[s]*old[s][t]   and   S_new[s][t] = td[s]*old + k[s]*v[t]
  const float vt = sh_v[t];
  float rs = 0.0f;
  for (int s = sg * 16; s < sg * 16 + 16; ++s) {
    const float o = ob[s * S_ + t];       // coalesced across t
    rs += sh_r[s] * o;
    nb[s * S_ + t] = sh_td[s] * o + sh_k[s] * vt;
  }
  part[sg][t] = rs;
  __syncthreads();
  if (threadIdx.x < 64) {
    wkv_s[t] = SCALE_ * (da * vt + part[0][t] + part[1][t] + part[2][t] + part[3][t]);
  }
  __syncthreads();

  // group norm over the 64 columns of this head
  const float wv = (threadIdx.x < 64) ? wkv_s[t] : 0.0f;
  const float mean = block_sum256(wv, red) * (1.0f / S_);
  const float ct = (threadIdx.x < 64) ? (wv - mean) : 0.0f;
  const float var = block_sum256(ct * ct, red) * (1.0f / S_);
  if (threadIdx.x < 64) {
    const int e = h * S_ + t;
    const float yy = ct * rsqrtf(var + GN_EPS_) * lnxw[e] + lnxb[e];
    yg[e] = yy * gv[e];
  }
}

// ---------------- final LN (1 block) ----------------
__global__ __launch_bounds__(256) void final_ln(
    const float* __restrict__ x, const float* __restrict__ lw,
    const float* __restrict__ lb, float* __restrict__ xf) {
  __shared__ float s8[8];
  float lv[4]; float sum = 0.0f;
#pragma unroll
  for (int j = 0; j < 4; ++j) { lv[j] = x[threadIdx.x + j * 256]; sum += lv[j]; }
  const float mean = block_sum256(sum, s8) * (1.0f / E_);
  float vs = 0.0f;
#pragma unroll
  for (int j = 0; j < 4; ++j) { float t = lv[j] - mean; vs += t * t; }
  const float rstd = rsqrtf(block_sum256(vs, s8) * (1.0f / E_) + 1e-5f);
#pragma unroll
  for (int j = 0; j < 4; ++j) {
    const int i = threadIdx.x + j * 256;
    xf[i] = (lv[j] - mean) * rstd * lw[i] + lb[i];
  }
}

// ---------------- host-side launcher ----------------
extern "C" void kernel_launch(void* const* d_in, const int* in_sizes, int n_in,
                              void* d_out, int out_size, void* d_ws, size_t ws_size,
                              hipStream_t stream) {
  (void)in_sizes; (void)n_in; (void)out_size; (void)ws_size;

  const int*   token  = (const int*)d_in[0];
  const float* s_att  = (const float*)d_in[1];   // (L,1,E)
  const float* s_wkv  = (const float*)d_in[2];   // (L,H,S,S)
  const float* s_ffn  = (const float*)d_in[3];   // (L,1,E)
  const float* emb_w  = (const float*)d_in[4];
  const float* ln0w   = (const float*)d_in[5];
  const float* ln0b   = (const float*)d_in[6];
  const float* ln1w   = (const float*)d_in[7];
  const float* ln1b   = (const float*)d_in[8];
  const float* ln2w   = (const float*)d_in[9];
  const float* ln2b   = (const float*)d_in[10];
  const float* tmk    = (const float*)d_in[11];
  const float* tmv    = (const float*)d_in[12];
  const float* tmr    = (const float*)d_in[13];
  const float* tmg    = (const float*)d_in[14];
  const float* tfirst = (const float*)d_in[15];  // (L,H,S,1) -> L*1024
  const float* tdecay = (const float*)d_in[16];
  const float* att_kw = (const float*)d_in[17];
  const float* att_vw = (const float*)d_in[18];
  const float* att_rw = (const float*)d_in[19];
  const float* att_gw = (const float*)d_in[20];
  const float* att_ow = (const float*)d_in[21];
  const float* lnxw   = (const float*)d_in[22];
  const float* lnxb   = (const float*)d_in[23];
  const float* ftmk   = (const float*)d_in[24];
  const float* ftmr   = (const float*)d_in[25];
  const float* ffn_kw = (const float*)d_in[26];  // (L,FF,E)
  const float* ffn_vw = (const float*)d_in[27];  // (L,E,FF)
  const float* ffn_rw = (const float*)d_in[28];  // (L,E,E)
  const float* lnow   = (const float*)d_in[29];
  const float* lnob   = (const float*)d_in[30];
  const float* head_w = (const float*)d_in[31];

  // workspace layout (floats)
  float* ws   = (float*)d_ws;
  float* x    = ws;              // E
  float* xk   = ws + 1024;       // E
  float* xv   = ws + 2048;
  float* xr   = ws + 3072;
  float* xg   = ws + 4096;
  float* rv   = ws + 5120;
  float* kvv  = ws + 6144;
  float* vvv  = ws + 7168;
  float* gvv  = ws + 8192;
  float* yg   = ws + 9216;
  float* kk   = ws + 10240;      // FF
  float* rr   = ws + 13824;      // E
  float* xk2  = ws + 14848;
  float* xr2  = ws + 15872;
  float* xf   = ws + 16896;

  // output layout: logits | new_as | new_wk | new_fs
  float* out    = (float*)d_out;
  float* logits = out;
  float* new_as = out + V_;
  float* new_wk = new_as + L_ * E_;
  float* new_fs = new_wk + (size_t)L_ * H_ * S_ * S_;

  const size_t EE = (size_t)E_ * E_;
  const size_t FE = (size_t)FF_ * E_;

  embed_ln<<<1, 256, 0, stream>>>(token, emb_w, ln0w, ln0b, x);

  for (int i = 0; i < L_; ++i) {
    att_pre<<<1, 256, 0, stream>>>(x, ln1w + i * E_, ln1b + i * E_, s_att + i * E_,
                                   tmk + i * E_, tmv + i * E_, tmr + i * E_, tmg + i * E_,
                                   xk, xv, xr, xg, new_as + i * E_);
    matvec_wmma<<<8, 256, 0, stream>>>(att_rw + i * EE, xr, rv,  nullptr, nullptr, E_, E_, 0);
    matvec_wmma<<<8, 256, 0, stream>>>(att_kw + i * EE, xk, kvv, nullptr, nullptr, E_, E_, 0);
    matvec_wmma<<<8, 256, 0, stream>>>(att_vw + i * EE, xv, vvv, nullptr, nullptr, E_, E_, 0);
    matvec_wmma<<<8, 256, 0, stream>>>(att_gw + i * EE, xg, gvv, nullptr, nullptr, E_, E_, 1);
    wkv_gn<<<H_, 256, 0, stream>>>(rv, kvv, vvv, gvv, tfirst + i * (H_ * S_),
                                   tdecay + i * (H_ * S_), s_wkv + (size_t)i * H_ * S_ * S_,
                                   lnxw + i * E_, lnxb + i * E_,
                                   new_wk + (size_t)i * H_ * S_ * S_, yg);
    matvec_wmma<<<8, 256, 0, stream>>>(att_ow + i * EE, yg, x, x, nullptr, E_, E_, 0);

    ffn_pre<<<1, 256, 0, stream>>>(x, ln2w + i * E_, ln2b + i * E_, s_ffn + i * E_,
                                   ftmk + i * E_, ftmr + i * E_, xk2, xr2, new_fs + i * E_);
    matvec_wmma<<<28, 256, 0, stream>>>(ffn_kw + i * FE, xk2, kk, nullptr, nullptr, FF_, E_, 3);
    matvec_wmma<<<8,  256, 0, stream>>>(ffn_rw + i * EE, xr2, rr, nullptr, nullptr, E_, E_, 2);
    matvec_wmma<<<8,  256, 0, stream>>>(ffn_vw + i * FE, kk, x, x, rr, E_, FF_, 4);
  }

  final_ln<<<1, 256, 0, stream>>>(x, lnow, lnob, xf);
  matvec_wmma<<<393, 256, 0, stream>>>(head_w, xf, logits, nullptr, nullptr, V_, E_, 0);
}